// AggregationMPNN_18365280157752
// MI455X (gfx1250) — compile-verified
//
#include <hip/hip_runtime.h>

// CDNA5 / gfx1250 MPNN: f32 WMMA (V_WMMA_F32_16X16X4_F32), one workgroup per graph.
// LDS row strides padded (+4 floats) so A-fragment column reads hit distinct banks.
typedef __attribute__((ext_vector_type(2))) float v2f;
typedef __attribute__((ext_vector_type(8))) float v8f;

#define BNODES 64
#define FDIM   64
#define EDIM   16
#define MDIM   128
#define ODIM   128
#define PASSES 4
#define NTHR   256
#define LD64   68    // padded stride for 64-wide LDS tiles (bank-conflict-free columns)
#define LD128  132   // padded stride for 128-wide LDS tiles

__device__ __forceinline__ v8f wmma_f32(v2f a, v2f b, v8f c) {
  // D = A(16x4) * B(4x16) + C(16x16), fp32
  return __builtin_amdgcn_wmma_f32_16x16x4_f32(false, a, false, b, (short)0, c, false, false);
}

__device__ __forceinline__ v2f lds_a(const float* __restrict__ A, int lda, int row, int kb) {
  const float* p = A + row * lda + kb;
  v2f a; a[0] = p[0]; a[1] = p[1];
  return a;
}
__device__ __forceinline__ v2f mat_b(const float* __restrict__ B, int ldb, int kb, int col) {
  const float* p = B + kb * ldb + col;
  v2f v; v[0] = p[0]; v[1] = p[ldb];
  return v;
}

__global__ __launch_bounds__(NTHR)
void mpnn_kernel(const int*   __restrict__ adjacency,
                 const float* __restrict__ nodes,
                 const float* __restrict__ edges,
                 const float* __restrict__ W_n,
                 const float* __restrict__ W_e,
                 const float* __restrict__ b_m,
                 const float* __restrict__ W_u,
                 const float* __restrict__ b_u,
                 const float* __restrict__ W_r,
                 const float* __restrict__ b_r,
                 float*       __restrict__ out)
{
  extern __shared__ float smem[];
  float* adjf   = smem;                        // 64 x LD64
  float* hn     = adjf   + BNODES * LD64;      // 64 x LD64 original node features
  float* buf0   = hn     + BNODES * LD64;      // 64 x LD64 h (ping)
  float* buf1   = buf0   + BNODES * LD64;      // 64 x LD64 agg scratch
  float* buf2   = buf1   + BNODES * LD64;      // 64 x LD64 h (pong)
  float* msg    = buf2   + BNODES * LD64;      // 64 x LD128
  float* base   = msg    + BNODES * LD128;     // 64 x LD128 (agg_e @ W_e + b_m, pass-invariant)
  float* aggE   = base   + BNODES * LD128;     // 64 x 16
  float* maskF  = aggE   + BNODES * EDIM;      // 64
  float* outAcc = maskF  + BNODES;             // 128

  const int b    = blockIdx.x;
  const int tid  = threadIdx.x;
  const int lane = tid & 31;
  const int wave = tid >> 5;
  const int cl   = lane & 15;   // column / row-in-tile index
  const int half = lane >> 4;   // 0 or 1

  const size_t adjBase  = (size_t)b * BNODES * BNODES;
  const size_t nodeBase = (size_t)b * BNODES * FDIM;
  const size_t edgeBase = (size_t)b * BNODES * BNODES * EDIM;

  // Prefetch this graph's 256KB edge block (global_prefetch_b8); it is read exactly once.
  __builtin_prefetch(edges + edgeBase + (size_t)tid * 256, 0, 1);

  // ---- stage adjacency (as float) and nodes into LDS (padded strides) ----
  for (int i = tid; i < BNODES * BNODES; i += NTHR) {
    const int r = i >> 6, c = i & 63;
    adjf[r * LD64 + c] = (float)adjacency[adjBase + i];
  }
  for (int i = tid; i < BNODES * FDIM; i += NTHR) {
    const int r = i >> 6, c = i & 63;
    const float v = nodes[nodeBase + i];
    hn[r * LD64 + c]   = v;
    buf0[r * LD64 + c] = v;
  }
  if (tid < ODIM) outAcc[tid] = 0.0f;
  __syncthreads();

  // ---- degree mask ----
  if (tid < BNODES) {
    float s = 0.f;
    const float* row = adjf + tid * LD64;
    #pragma unroll 8
    for (int u = 0; u < BNODES; ++u) s += row[u];
    maskF[tid] = (s > 0.5f) ? 1.0f : 0.0f;
  }

  // ---- agg_e[v][e] = sum_u adj[v][u] * edges[v][u][e]  (single streaming pass) ----
  {
    const int v  = tid >> 2;
    const int eg = (tid & 3) * 4;
    const float* ep   = edges + edgeBase + (size_t)v * BNODES * EDIM + eg;
    const float* arow = adjf + v * LD64;
    float s0 = 0.f, s1 = 0.f, s2 = 0.f, s3 = 0.f;
    for (int u = 0; u < BNODES; ++u) {
      const float a = arow[u];
      const float* e = ep + u * EDIM;
      s0 += a * e[0]; s1 += a * e[1]; s2 += a * e[2]; s3 += a * e[3];
    }
    float* d = aggE + v * EDIM + eg;
    d[0] = s0; d[1] = s1; d[2] = s2; d[3] = s3;
  }
  __syncthreads();

  // ---- base_msg = agg_e @ W_e + b_m (pass-invariant, tiny K=16) ----
  for (int i = tid; i < BNODES * MDIM; i += NTHR) {
    const int v = i >> 7, m = i & 127;
    float s = b_m[m];
    const float* ae = aggE + v * EDIM;
    #pragma unroll
    for (int e = 0; e < EDIM; ++e) s += ae[e] * W_e[e * MDIM + m];
    base[v * LD128 + m] = s;
  }
  __syncthreads();

  float* hCur = buf0;
  float* hNew = buf2;
  float* agg  = buf1;

  for (int pass = 0; pass < PASSES; ++pass) {
    // ---- A: agg(64x64) = adjf @ hCur ; 16 tiles, 2 per wave ----
    for (int t = wave * 2; t < wave * 2 + 2; ++t) {
      const int mT = t >> 2, nT = t & 3;
      v8f acc;
      #pragma unroll
      for (int r = 0; r < 8; ++r) acc[r] = 0.f;
      #pragma unroll 4
      for (int kk = 0; kk < 16; ++kk) {
        const int kb = kk * 4 + half * 2;
        acc = wmma_f32(lds_a(adjf, LD64, mT * 16 + cl, kb),
                       mat_b(hCur, LD64, kb, nT * 16 + cl), acc);
      }
      float* d = agg + (mT * 16 + half * 8) * LD64 + nT * 16 + cl;
      #pragma unroll
      for (int r = 0; r < 8; ++r) d[r * LD64] = acc[r];
    }
    __syncthreads();

    // ---- B: msg(64x128) = agg @ W_n + base ; 32 tiles, 4 per wave ----
    for (int i = 0; i < 4; ++i) {
      const int t = wave * 4 + i;
      const int mT = t >> 3, nT = t & 7;
      v8f acc;
      const float* bp = base + (mT * 16 + half * 8) * LD128 + nT * 16 + cl;
      #pragma unroll
      for (int r = 0; r < 8; ++r) acc[r] = bp[r * LD128];
      #pragma unroll 4
      for (int kk = 0; kk < 16; ++kk) {
        const int kb = kk * 4 + half * 2;
        acc = wmma_f32(lds_a(agg, LD64, mT * 16 + cl, kb),
                       mat_b(W_n, MDIM, kb, nT * 16 + cl), acc);
      }
      float* d = msg + (mT * 16 + half * 8) * LD128 + nT * 16 + cl;
      #pragma unroll
      for (int r = 0; r < 8; ++r) d[r * LD128] = acc[r];
    }
    __syncthreads();

    // ---- C: hNew = relu(h @ Wu[0:64] + msg @ Wu[64:192] + b_u), masked ----
    for (int t = wave * 2; t < wave * 2 + 2; ++t) {
      const int mT = t >> 2, nT = t & 3;
      const float bu = b_u[nT * 16 + cl];
      v8f acc;
      #pragma unroll
      for (int r = 0; r < 8; ++r) acc[r] = bu;
      #pragma unroll 4
      for (int kk = 0; kk < 16; ++kk) {
        const int kb = kk * 4 + half * 2;
        acc = wmma_f32(lds_a(hCur, LD64, mT * 16 + cl, kb),
                       mat_b(W_u, FDIM, kb, nT * 16 + cl), acc);
      }
      #pragma unroll 4
      for (int kk = 0; kk < 32; ++kk) {
        const int kb = kk * 4 + half * 2;
        acc = wmma_f32(lds_a(msg, LD128, mT * 16 + cl, kb),
                       mat_b(W_u + 64 * FDIM, FDIM, kb, nT * 16 + cl), acc);
      }
      #pragma unroll
      for (int r = 0; r < 8; ++r) {
        const int m = mT * 16 + half * 8 + r;
        const int c = nT * 16 + cl;
        const float nv = fmaxf(acc[r], 0.f);
        hNew[m * LD64 + c] = (maskF[m] > 0.5f) ? nv : hCur[m * LD64 + c];
      }
    }
    __syncthreads();
    float* tp = hCur; hCur = hNew; hNew = tp;
  }

  // ---- readout: r = relu(h @ Wr[0:64] + nodes @ Wr[64:128] + b_r); masked column sum ----
  for (int i = 0; i < 4; ++i) {
    const int t = wave * 4 + i;
    const int mT = t >> 3, nT = t & 7;
    const float br = b_r[nT * 16 + cl];
    v8f acc;
    #pragma unroll
    for (int r = 0; r < 8; ++r) acc[r] = br;
    #pragma unroll 4
    for (int kk = 0; kk < 16; ++kk) {
      const int kb = kk * 4 + half * 2;
      acc = wmma_f32(lds_a(hCur, LD64, mT * 16 + cl, kb),
                     mat_b(W_r, ODIM, kb, nT * 16 + cl), acc);
    }
    #pragma unroll 4
    for (int kk = 0; kk < 16; ++kk) {
      const int kb = kk * 4 + half * 2;
      acc = wmma_f32(lds_a(hn, LD64, mT * 16 + cl, kb),
                     mat_b(W_r + 64 * ODIM, ODIM, kb, nT * 16 + cl), acc);
    }
    float ps = 0.f;
    #pragma unroll
    for (int r = 0; r < 8; ++r) {
      const int m = mT * 16 + half * 8 + r;
      ps += maskF[m] * fmaxf(acc[r], 0.f);
    }
    atomicAdd(&outAcc[nT * 16 + cl], ps);   // ds_add_f32
  }
  __syncthreads();
  if (tid < ODIM) out[(size_t)b * ODIM + tid] = outAcc[tid];
}

extern "C" void kernel_launch(void* const* d_in, const int* in_sizes, int n_in,
                              void* d_out, int out_size, void* d_ws, size_t ws_size,
                              hipStream_t stream) {
  const int*   adjacency = (const int*)  d_in[0];
  const float* nodes     = (const float*)d_in[1];
  const float* edges     = (const float*)d_in[2];
  const float* W_n       = (const float*)d_in[3];
  const float* W_e       = (const float*)d_in[4];
  const float* b_m       = (const float*)d_in[5];
  const float* W_u       = (const float*)d_in[6];
  const float* b_u       = (const float*)d_in[7];
  const float* W_r       = (const float*)d_in[8];
  const float* b_r       = (const float*)d_in[9];
  float* out = (float*)d_out;

  const int B = in_sizes[0] / (BNODES * BNODES);
  const size_t lds_floats =
      5u * BNODES * LD64 + 2u * BNODES * LD128 + BNODES * EDIM + BNODES + ODIM;
  const size_t lds_bytes = lds_floats * sizeof(float);

  hipLaunchKernelGGL(mpnn_kernel, dim3(B), dim3(NTHR), lds_bytes, stream,
                     adjacency, nodes, edges, W_n, W_e, b_m, W_u, b_u, W_r, b_r, out);
}